// Attention_13005160972550
// MI455X (gfx1250) — compile-verified
//
#include <hip/hip_runtime.h>
#include <hip/hip_bf16.h>
#include <math.h>

// Problem constants (from the reference)
#define BB 4
#define QQ 256
#define CC 1024
#define HH 128
#define NEG_SLOPE 0.01f

// score-kernel tiling
#define QT   8      // queries per block
#define CCH  64     // context rows staged per chunk
#define HSTR 132    // padded LDS row stride (floats): 132%64==4 -> conflict-free row sweeps

typedef __attribute__((ext_vector_type(2))) float v2f;
typedef __attribute__((ext_vector_type(8))) float v8f;

// D(16x16,f32) = A(16x4,f32) * B(4x16,f32) + C   — full-precision matrix pipe
__device__ __forceinline__ v8f wmma_f32_16x16x4(v2f a, v2f b, v8f c) {
    return __builtin_amdgcn_wmma_f32_16x16x4_f32(
        /*neg_a=*/false, a, /*neg_b=*/false, b,
        /*c_mod=*/(short)0, c, /*reuse_a=*/false, /*reuse_b=*/false);
}

// ---------------------------------------------------------------------------
// dst[row, n] = sum_k src[row, k] * Ww[n, wcol_off + k]  (+ bias[n])
// One wave per 16x16 tile; K-loop over 128 in steps of 4 f32-WMMAs.
// ---------------------------------------------------------------------------
__global__ __launch_bounds__(128) void proj_kernel(
    const float* __restrict__ src, const float* __restrict__ Ww,
    const float* __restrict__ bias, float* __restrict__ dst,
    int nrows, int wcol_off)
{
    const int lane = threadIdx.x & 31;
    const int wave = threadIdx.x >> 5;
    const int tile = blockIdx.x * (blockDim.x >> 5) + wave;
    const int mt = tile >> 3;          // 8 N-tiles (128/16)
    const int nt = tile & 7;
    if (mt * 16 >= nrows) return;      // wave-uniform -> EXEC stays all-ones

    const int m  = lane & 15;                // A: lane -> M row
    const int kh = (lane >> 4) * 2;          // lanes 16-31 carry K=2,3
    const int n  = nt * 16 + (lane & 15);    // B/C/D: lane -> N col

    v8f acc;
    const float binit = bias ? bias[n] : 0.0f;
#pragma unroll
    for (int i = 0; i < 8; ++i) acc[i] = binit;

    const float* arow = src + (mt * 16 + m) * HH;
    const float* bcol = Ww + n * (2 * HH) + wcol_off;   // B[k][n] = Ww[n, off+k]
#pragma unroll 4
    for (int k = 0; k < HH; k += 4) {
        v2f a, bm;
        a.x  = arow[k + kh];
        a.y  = arow[k + kh + 1];
        bm.x = bcol[k + kh];
        bm.y = bcol[k + kh + 1];
        acc = wmma_f32_16x16x4(a, bm, acc);
    }

    const int rbase = mt * 16 + ((lane >> 4) ? 8 : 0);
#pragma unroll
    for (int r = 0; r < 8; ++r)
        dst[(rbase + r) * HH + n] = acc[r];
}

// ---------------------------------------------------------------------------
// Raw scores: score[b,q,c] = sum_k Ws[k]*leaky(hq[b,q,k]+hc[b,c,k]) + bs
// Block = 256 threads handles QT=8 queries; hc staged per 64-row chunk into
// padded LDS (conflict-free b128 row reads). Thread owns one c + two q's:
// serial k-reduction in registers -> zero cross-lane reduction traffic.
// hc L2 traffic: 512MB -> 64MB vs. the naive per-(b,q) mapping.
// ---------------------------------------------------------------------------
__global__ __launch_bounds__(256) void score_kernel(
    const float* __restrict__ hq, const float* __restrict__ hc,
    const float* __restrict__ Ws, const float* __restrict__ bs,
    float* __restrict__ score)        // (B,Q,C) raw scores (attn slot of d_out)
{
    __shared__ float s_hc[CCH * HSTR];   // 33792 B
    __shared__ float s_hq[QT * HSTR];    //  4224 B
    __shared__ float s_ws[HH];           //   512 B

    const int tid   = threadIdx.x;
    const int blk   = blockIdx.x;              // 0 .. B*QQ/QT-1 = 127
    const int b     = blk / (QQ / QT);
    const int qbase = (blk % (QQ / QT)) * QT;

    // stage hq tile (8 x 128) and Ws (128) once per block
    {
        const int row = tid >> 5, col4 = (tid & 31) * 4;   // 256 float4 = 1024 f
        *(float4*)&s_hq[row * HSTR + col4] =
            *(const float4*)&hq[(b * QQ + qbase + row) * HH + col4];
        if (tid < 32)
            *(float4*)&s_ws[tid * 4] = *(const float4*)&Ws[tid * 4];
    }
    const float bsv = bs[0];

    const int c  = tid & 63;       // context row within chunk
    const int qg = tid >> 6;       // 0..3 -> q pair {2qg, 2qg+1}
    const int q0 = qg * 2, q1 = q0 + 1;
    const float* hcb = hc + (size_t)b * CC * HH;

    for (int cb = 0; cb < CC; cb += CCH) {
        __syncthreads();           // also covers hq/Ws staging on 1st iter
        // stage 64 x 128 f32 (2048 float4), 8 float4 per thread, coalesced
#pragma unroll
        for (int j = 0; j < 8; ++j) {
            const int idx = tid + j * 256;
            const int row = idx >> 5, col4 = (idx & 31) * 4;
            *(float4*)&s_hc[row * HSTR + col4] =
                *(const float4*)&hcb[(cb + row) * HH + col4];
        }
        __syncthreads();

        float acc0 = 0.0f, acc1 = 0.0f;
        const float* hcrow = &s_hc[c * HSTR];
#pragma unroll 8
        for (int k = 0; k < HH; k += 4) {
            const float4 hc4 = *(const float4*)&hcrow[k];              // b128, conflict-free
            const float4 a4  = *(const float4*)&s_hq[q0 * HSTR + k];   // broadcast
            const float4 b4  = *(const float4*)&s_hq[q1 * HSTR + k];   // broadcast
            const float4 w4  = *(const float4*)&s_ws[k];               // broadcast
            float x;
            x = a4.x + hc4.x; acc0 += w4.x * fmaxf(x, NEG_SLOPE * x);
            x = a4.y + hc4.y; acc0 += w4.y * fmaxf(x, NEG_SLOPE * x);
            x = a4.z + hc4.z; acc0 += w4.z * fmaxf(x, NEG_SLOPE * x);
            x = a4.w + hc4.w; acc0 += w4.w * fmaxf(x, NEG_SLOPE * x);
            x = b4.x + hc4.x; acc1 += w4.x * fmaxf(x, NEG_SLOPE * x);
            x = b4.y + hc4.y; acc1 += w4.y * fmaxf(x, NEG_SLOPE * x);
            x = b4.z + hc4.z; acc1 += w4.z * fmaxf(x, NEG_SLOPE * x);
            x = b4.w + hc4.w; acc1 += w4.w * fmaxf(x, NEG_SLOPE * x);
        }
        score[(size_t)(b * QQ + qbase + q0) * CC + cb + c] = acc0 + bsv;
        score[(size_t)(b * QQ + qbase + q1) * CC + cb + c] = acc1 + bsv;
    }
}

// ---------------------------------------------------------------------------
// In-place softmax over c for each (b,q) row of the attn buffer.
// ---------------------------------------------------------------------------
__global__ __launch_bounds__(256) void softmax_kernel(float* __restrict__ attn)
{
    __shared__ float s[CC];
    __shared__ float red[8];

    const int bq   = blockIdx.x;
    const int tid  = threadIdx.x;
    const int lane = tid & 31;
    const int wave = tid >> 5;
    float* row = attn + (size_t)bq * CC;

#pragma unroll
    for (int j = 0; j < 4; ++j)
        s[tid + j * 256] = row[tid + j * 256];
    __syncthreads();

    float m = -__builtin_inff();
#pragma unroll
    for (int j = 0; j < 4; ++j)
        m = fmaxf(m, s[tid + j * 256]);
#pragma unroll
    for (int off = 16; off > 0; off >>= 1)
        m = fmaxf(m, __shfl_xor(m, off, 32));
    if (lane == 0) red[wave] = m;
    __syncthreads();
    m = red[0];
#pragma unroll
    for (int w = 1; w < 8; ++w) m = fmaxf(m, red[w]);

    float psum = 0.0f;
#pragma unroll
    for (int j = 0; j < 4; ++j) {
        const int i = tid + j * 256;
        const float p = __expf(s[i] - m);
        s[i] = p;
        psum += p;
    }
#pragma unroll
    for (int off = 16; off > 0; off >>= 1)
        psum += __shfl_xor(psum, off, 32);
    __syncthreads();                   // all reads of red[] (max) done
    if (lane == 0) red[wave] = psum;
    __syncthreads();
    float sum = 0.0f;
#pragma unroll
    for (int w = 0; w < 8; ++w) sum += red[w];
    const float inv = 1.0f / sum;
#pragma unroll
    for (int j = 0; j < 4; ++j) {
        const int i = tid + j * 256;
        row[i] = s[i] * inv;
    }
}

// ---------------------------------------------------------------------------
// attn_output[b,q,h] = sum_c attn[b,q,c] * context[b,c,h]
// Batched GEMM M=256, N=128, K=1024 per b; 256 f32-WMMAs per 16x16 tile.
// ---------------------------------------------------------------------------
__global__ __launch_bounds__(128) void attn_out_kernel(
    const float* __restrict__ attn, const float* __restrict__ context,
    float* __restrict__ out)
{
    const int lane = threadIdx.x & 31;
    const int wave = threadIdx.x >> 5;
    const int tile = blockIdx.x * (blockDim.x >> 5) + wave;  // 0..511
    const int b   = tile >> 7;
    const int rem = tile & 127;
    const int mt  = rem >> 3;
    const int nt  = rem & 7;

    const int m  = lane & 15;
    const int kh = (lane >> 4) * 2;
    const int n  = nt * 16 + (lane & 15);

    v8f acc = {};
    const float* arow = attn + (b * QQ + mt * 16 + m) * CC;
    const float* bcol = context + b * CC * HH + n;
#pragma unroll 2
    for (int k = 0; k < CC; k += 4) {
        v2f a, bm;
        a.x  = arow[k + kh];
        a.y  = arow[k + kh + 1];
        bm.x = bcol[(k + kh) * HH];
        bm.y = bcol[(k + kh + 1) * HH];
        acc = wmma_f32_16x16x4(a, bm, acc);
    }

    const int rbase = b * QQ + mt * 16 + ((lane >> 4) ? 8 : 0);
#pragma unroll
    for (int r = 0; r < 8; ++r)
        out[(rbase + r) * HH + n] = acc[r];
}

extern "C" void kernel_launch(void* const* d_in, const int* in_sizes, int n_in,
                              void* d_out, int out_size, void* d_ws, size_t ws_size,
                              hipStream_t stream) {
    (void)in_sizes; (void)n_in; (void)out_size; (void)ws_size;

    const float* query   = (const float*)d_in[0];   // (B,Q,H)
    const float* context = (const float*)d_in[1];   // (B,C,H)
    const float* Ww      = (const float*)d_in[2];   // (H,2H)
    const float* bw      = (const float*)d_in[3];   // (H)
    const float* Ws      = (const float*)d_in[4];   // (1,H)
    const float* bs      = (const float*)d_in[5];   // (1)

    float* out      = (float*)d_out;
    float* attn_out = out;                           // (B,Q,H)
    float* attn     = out + (size_t)BB * QQ * HH;    // (B,Q,C)

    float* hq = (float*)d_ws;                        // (B*Q,H)  512 KB
    float* hc = hq + (size_t)BB * QQ * HH;           // (B*C,H)  2 MB

    // hq = query @ Wq^T        : 512 wave-tiles
    proj_kernel<<<128, 128, 0, stream>>>(query, Ww, nullptr, hq, BB * QQ, 0);
    // hc = context @ Wc^T + bw : 2048 wave-tiles
    proj_kernel<<<512, 128, 0, stream>>>(context, Ww, bw, hc, BB * CC, HH);
    // raw scores -> attn slot of d_out (q-blocked, LDS-staged)
    score_kernel<<<BB * QQ / QT, 256, 0, stream>>>(hq, hc, Ws, bs, attn);
    // in-place softmax per (b,q) row
    softmax_kernel<<<BB * QQ, 256, 0, stream>>>(attn);
    // attn_output = attn @ context
    attn_out_kernel<<<128, 128, 0, stream>>>(attn, context, attn_out);
}